// SentAttNet_901943132714
// MI455X (gfx1250) — compile-verified
//
#include <hip/hip_runtime.h>
#include <cstdint>

// Problem geometry (fixed by the reference): B=4096, S=2048, F=4.
#define S_LEN 2048
#define BLOCK 256
#define NCH (S_LEN / BLOCK)    // 8 chunks of 256 s-positions per row
#define WAVE 32                // gfx1250 is wave32-only
#define NWAVES (BLOCK / WAVE)  // 8 waves per workgroup

#if defined(__has_builtin)
#if __has_builtin(__builtin_amdgcn_global_load_async_to_lds_b128)
#define HAVE_ASYNC_B128_BUILTIN 1
#endif
#if __has_builtin(__builtin_amdgcn_s_wait_asynccnt)
#define HAVE_WAIT_ASYNC_BUILTIN 1
#endif
#endif

typedef int v4i __attribute__((ext_vector_type(4)));
typedef __attribute__((address_space(1))) v4i* gptr_b128;  // global int4*
typedef __attribute__((address_space(3))) v4i* lptr_b128;  // LDS int4*
typedef __attribute__((address_space(3))) void* lptr_void;

// Async DMA of 16 bytes per lane: global -> LDS, tracked by ASYNCcnt.
__device__ __forceinline__ static void async_ld_b128(const void* g, void* l) {
#if defined(HAVE_ASYNC_B128_BUILTIN)
  __builtin_amdgcn_global_load_async_to_lds_b128((gptr_b128)g, (lptr_b128)l,
                                                 /*offset=*/0, /*cpol=*/0);
#else
  unsigned lds_off = (unsigned)(uintptr_t)(lptr_void)l;  // 32-bit LDS byte address
  asm volatile("global_load_async_to_lds_b128 %0, %1, off"
               :: "v"(lds_off), "v"(g) : "memory");
#endif
}

template <int N>
__device__ __forceinline__ static void wait_asynccnt() {
#if defined(HAVE_WAIT_ASYNC_BUILTIN)
  __builtin_amdgcn_s_wait_asynccnt(N);
#else
  asm volatile("s_wait_asynccnt %0" :: "i"(N) : "memory");
#endif
}

// Branch-free tanh: one v_exp_f32 + one v_rcp_f32, v_cndmask for tiny |z|
// (avoids the cancellation in (1-t)/(1+t) as t->1), sign via copysign (v_bfi).
// Keeps the hot loop free of exec-mask branches, unlike OCML tanhf.
__device__ __forceinline__ static float fast_tanhf(float z) {
  const float a = fabsf(z);
  const float t = __expf(-2.0f * a);                        // e^(-2a)
  const float r = (1.0f - t) * __builtin_amdgcn_rcpf(1.0f + t);
  const float p = fmaf(-0.33333333f * a, a * a, a);         // a - a^3/3
  const float sel = (a < 0.0625f) ? p : r;
  return copysignf(sel, z);
}

__global__ __launch_bounds__(BLOCK) void sentatt_softmax_kernel(
    const float* __restrict__ x,     // [B, S, 4]
    const float* __restrict__ w,     // [4, 1]
    const float* __restrict__ bias,  // [1]
    float* __restrict__ out)         // [B, S]
{
  // Double-buffered staging: 2 x (256 lanes * 16B) = 8 KB of the 320 KB WGP LDS.
  __shared__ __align__(16) float stage[2 * BLOCK * 4];
  __shared__ float red_max[NWAVES];
  __shared__ float red_sum[NWAVES];

  const int t = threadIdx.x;
  const int b = blockIdx.x;

  const float w0 = w[0], w1 = w[1], w2 = w[2], w3 = w[3];
  const float bs = bias[0];

  const float* xrow = x + (size_t)b * (S_LEN * 4);

  // ---- Pass 1: stream x through LDS, compute scores, track row max ----
  async_ld_b128(xrow + t * 4, &stage[t * 4]);  // prologue: chunk 0

  float sc[NCH];
  float vmax = -3.402823466e+38f;

#pragma unroll
  for (int i = 0; i < NCH; ++i) {
    if (i + 1 < NCH) {
      // Prefetch next chunk into the other buffer while we consume this one.
      async_ld_b128(xrow + ((size_t)(i + 1) * BLOCK + t) * 4,
                    &stage[((i + 1) & 1) * (BLOCK * 4) + t * 4]);
      wait_asynccnt<1>();  // chunk i complete (async loads retire in order)
    } else {
      wait_asynccnt<0>();
    }
    // Each wave reads exactly the LDS bytes its own lanes DMA'd: no barrier needed.
    const float4 v =
        *reinterpret_cast<const float4*>(&stage[(i & 1) * (BLOCK * 4) + t * 4]);
    const float z = fmaf(v.x, w0, fmaf(v.y, w1, fmaf(v.z, w2, fmaf(v.w, w3, bs))));
    const float s = fast_tanhf(z);
    sc[i] = s;
    vmax = fmaxf(vmax, s);
  }

  // ---- Row max: wave32 butterfly, then cross-wave via LDS ----
#pragma unroll
  for (int d = WAVE / 2; d >= 1; d >>= 1)
    vmax = fmaxf(vmax, __shfl_xor(vmax, d, WAVE));
  if ((t & (WAVE - 1)) == 0) red_max[t / WAVE] = vmax;
  __syncthreads();
  float m = red_max[0];
#pragma unroll
  for (int j = 1; j < NWAVES; ++j) m = fmaxf(m, red_max[j]);

  // ---- Pass 2: masked exp + row sum ----
  float e[NCH];
  float acc = 0.0f;
#pragma unroll
  for (int i = 0; i < NCH; ++i) {
    const float ei = (sc[i] != 0.0f) ? __expf(sc[i] - m) : 0.0f;
    e[i] = ei;
    acc += ei;
  }
#pragma unroll
  for (int d = WAVE / 2; d >= 1; d >>= 1) acc += __shfl_xor(acc, d, WAVE);
  if ((t & (WAVE - 1)) == 0) red_sum[t / WAVE] = acc;
  __syncthreads();
  float tot = red_sum[0];
#pragma unroll
  for (int j = 1; j < NWAVES; ++j) tot += red_sum[j];
  const float inv = __builtin_amdgcn_rcpf(tot);

  // ---- Normalize + coalesced stores ----
  float* orow = out + (size_t)b * S_LEN;
#pragma unroll
  for (int i = 0; i < NCH; ++i) orow[i * BLOCK + t] = e[i] * inv;
}

extern "C" void kernel_launch(void* const* d_in, const int* in_sizes, int n_in,
                              void* d_out, int out_size, void* d_ws, size_t ws_size,
                              hipStream_t stream) {
  const float* x    = (const float*)d_in[0];  // [B, S, 4] fp32
  const float* w    = (const float*)d_in[1];  // [4, 1]    fp32
  const float* bias = (const float*)d_in[2];  // [1]       fp32
  float* out = (float*)d_out;                 // [B, S]    fp32

  const int B = out_size / S_LEN;  // 4096
  sentatt_softmax_kernel<<<dim3(B), dim3(BLOCK), 0, stream>>>(x, w, bias, out);
}